// Multihead_informer_25683904430307
// MI455X (gfx1250) — compile-verified
//
#include <hip/hip_runtime.h>
#include <math.h>

#define LL    2048
#define CC    128
#define BB    16
#define NTHR  256
#define PERT  8          // LL / NTHR
#define MAXU  64
#define SEP   132        // padded E row stride (floats)
#define LCH   128        // l-chunk per WMMA pass
#define EPSV  1e-5f

typedef float v2f  __attribute__((ext_vector_type(2)));
typedef float v8f  __attribute__((ext_vector_type(8)));
typedef unsigned int u32x4 __attribute__((ext_vector_type(4)));
typedef int   i32x4 __attribute__((ext_vector_type(4)));
typedef int   i32x8 __attribute__((ext_vector_type(8)));

__device__ __forceinline__ float wg_reduce_sum(float v, int lane, int wid, float* s8) {
  #pragma unroll
  for (int m = 16; m; m >>= 1) v += __shfl_xor(v, m, 32);
  if (lane == 0) s8[wid] = v;
  __syncthreads();
  float r = 0.f;
  #pragma unroll
  for (int w = 0; w < 8; ++w) r += s8[w];
  __syncthreads();
  return r;
}

__global__ __launch_bounds__(NTHR)
void informer_fused(const float* __restrict__ x,
                    const float* Wq, const float* bq,
                    const float* Wk, const float* bk,
                    const float* Wv, const float* bv,
                    const float* Wo, const float* bo,
                    const float* w1, const float* b1,
                    const float* w2, const float* b2,
                    const float* g1, const float* be1,
                    const float* g2, const float* be2,
                    const int* __restrict__ idx,
                    float* __restrict__ out,
                    int U, int NT)
{
  __shared__ float sX[LL];          // channel data x[b,:,c]
  __shared__ float sM[LL];          // sparsity measure, later ctx
  __shared__ float sE[16 * SEP];    // exp tile for WMMA
  __shared__ float sR[24];          // reduction scratch
  __shared__ float sRv[8];
  __shared__ int   sRi[8];
  __shared__ int   sSel[MAXU];
  __shared__ float sQs[MAXU];
  __shared__ float sMx[MAXU];
  __shared__ float sNum[MAXU];
  __shared__ float sDen[MAXU];

  const int tid  = threadIdx.x;
  const int lane = tid & 31;
  const int wid  = tid >> 5;
  const int blk  = blockIdx.x;
  const int c    = blk % CC;
  const int b    = blk / CC;
  const float* __restrict__ xb = x + (size_t)b * (LL * CC) + c;

  // ---------------- TDM: async strided-tile gather into LDS -----------------
  // Tile: tile_dim0=1 elem (contiguous), tile_dim1=2048 rows, row stride = C
  if (wid == 0) {
    unsigned long long ga = (unsigned long long)(uintptr_t)xb;
    unsigned int ldsoff   = (unsigned int)(uintptr_t)(&sX[0]);
    u32x4 g0;
    g0[0] = 1u;                                                 // count=1, user mode
    g0[1] = ldsoff;                                             // lds_addr
    g0[2] = (unsigned)(ga & 0xFFFFFFFFull);                     // global_addr lo
    g0[3] = (unsigned)((ga >> 32) & 0x1FFFFFFull) | (2u << 30); // addr hi | type=2
    i32x8 g1v;
    g1v[0] = (int)(2u << 16);             // workgroup_mask=0, data_size=4B
    g1v[1] = (int)(1u << 16);             // tensor_dim0 = 1 (bits 63:48)
    g1v[2] = (int)(((unsigned)LL) << 16); // tensor_dim1 lo16 (bits 95:80)
    g1v[3] = (int)(1u << 16);             // tensor_dim1 hi=0, tile_dim0=1
    g1v[4] = (int)LL;                     // tile_dim1=2048, tile_dim2=0
    g1v[5] = (int)CC;                     // tensor_dim0_stride = 128 elems
    g1v[6] = 0; g1v[7] = 0;
    i32x4 gz4 = {0, 0, 0, 0};
    i32x8 gz8 = {0, 0, 0, 0, 0, 0, 0, 0};
#if __has_builtin(__builtin_amdgcn_tensor_load_to_lds)
    __builtin_amdgcn_tensor_load_to_lds(g0, g1v, gz4, gz4, gz8, 0);
#endif
#if __has_builtin(__builtin_amdgcn_s_wait_tensorcnt)
    __builtin_amdgcn_s_wait_tensorcnt(0);
#endif
  }
  __syncthreads();
  // Redundant direct fill (correctness insurance; overwrites with same data)
  #pragma unroll
  for (int i = 0; i < PERT; ++i) {
    int l = tid + i * NTHR;
    sX[l] = xb[(size_t)l * CC];
  }
  __syncthreads();

  const float vWq = Wq[0], vbq = bq[0], vWk = Wk[0], vbk = bk[0];
  const float vWv = Wv[0], vbv = bv[0], vWo = Wo[0], vbo = bo[0];
  const float vw1 = w1[0], vb1 = b1[0], vw2 = w2[0], vb2 = b2[0];
  const float vg1 = g1[0], vbe1 = be1[0], vg2 = g2[0], vbe2 = be2[0];

  // ------------- pass 1: Σv, max(k), min(k) over the sequence --------------
  float vs = 0.f, kmx = -__builtin_inff(), kmn = __builtin_inff();
  #pragma unroll
  for (int i = 0; i < PERT; ++i) {
    float xv = sX[tid + i * NTHR];
    float kv = fmaf(xv, vWk, vbk);
    vs += fmaf(xv, vWv, vbv);
    kmx = fmaxf(kmx, kv);
    kmn = fminf(kmn, kv);
  }
  #pragma unroll
  for (int m = 16; m; m >>= 1) {
    vs  += __shfl_xor(vs, m, 32);
    kmx  = fmaxf(kmx, __shfl_xor(kmx, m, 32));
    kmn  = fminf(kmn, __shfl_xor(kmn, m, 32));
  }
  if (lane == 0) { sR[wid] = vs; sR[8 + wid] = kmx; sR[16 + wid] = kmn; }
  __syncthreads();
  float vsum = 0.f, kmaxA = -__builtin_inff(), kminA = __builtin_inff();
  #pragma unroll
  for (int w = 0; w < 8; ++w) {
    vsum  += sR[w];
    kmaxA  = fmaxf(kmaxA, sR[8 + w]);
    kminA  = fminf(kminA, sR[16 + w]);
  }
  __syncthreads();
  const float vmean = vsum / (float)LL;

  // ------------- sparsity measure M[l] via sampled-k gathers ---------------
  #pragma unroll
  for (int i = 0; i < PERT; ++i) {
    int l = tid + i * NTHR;
    float mx = -__builtin_inff(), mn = __builtin_inff(), sm = 0.f;
    if ((U & 3) == 0) {
      const int4* ip = (const int4*)(idx + (size_t)l * U);
      for (int j = 0; j < (U >> 2); ++j) {
        int4 id4 = ip[j];
        float k0 = fmaf(sX[id4.x], vWk, vbk);
        float k1 = fmaf(sX[id4.y], vWk, vbk);
        float k2 = fmaf(sX[id4.z], vWk, vbk);
        float k3 = fmaf(sX[id4.w], vWk, vbk);
        mx = fmaxf(fmaxf(mx, k0), fmaxf(k1, fmaxf(k2, k3)));
        mn = fminf(fminf(mn, k0), fminf(k1, fminf(k2, k3)));
        sm += (k0 + k1) + (k2 + k3);
      }
    } else {
      for (int j = 0; j < U; ++j) {
        float kg = fmaf(sX[idx[(size_t)l * U + j]], vWk, vbk);
        mx = fmaxf(mx, kg); mn = fminf(mn, kg); sm += kg;
      }
    }
    float q = fmaf(sX[l], vWq, vbq);
    sM[l] = q * (q >= 0.f ? mx : mn) - (q * sm) / (float)LL;
  }
  __syncthreads();

  // ------------- top-U selection (ties -> lower index, like lax.top_k) -----
  for (int it = 0; it < U; ++it) {
    float bvv = -__builtin_inff(); int bii = LL;
    #pragma unroll
    for (int i = 0; i < PERT; ++i) {
      int l = tid + i * NTHR;
      float mv = sM[l];
      if (mv > bvv || (mv == bvv && l < bii)) { bvv = mv; bii = l; }
    }
    #pragma unroll
    for (int m = 16; m; m >>= 1) {
      float ov = __shfl_xor(bvv, m, 32);
      int   oi = __shfl_xor(bii, m, 32);
      if (ov > bvv || (ov == bvv && oi < bii)) { bvv = ov; bii = oi; }
    }
    if (lane == 0) { sRv[wid] = bvv; sRi[wid] = bii; }
    __syncthreads();
    if (tid == 0) {
      float wv = sRv[0]; int wi = sRi[0];
      #pragma unroll
      for (int w = 1; w < 8; ++w)
        if (sRv[w] > wv || (sRv[w] == wv && sRi[w] < wi)) { wv = sRv[w]; wi = sRi[w]; }
      sSel[it] = wi;
      sM[wi]   = -__builtin_inff();
    }
    __syncthreads();
  }

  // per-selected-row q and softmax max (exact: rounding is monotone in k)
  if (tid < NT * 16) {
    int u  = tid;
    int lu = (u < U) ? sSel[u] : 0;
    float q = (u < U) ? fmaf(sX[lu], vWq, vbq) : 0.f;
    sQs[u] = q;
    sMx[u] = (q >= 0.f) ? q * kmaxA : q * kminA;
    if (u >= U) sSel[u] = 0;
  }
  __syncthreads();

  // ------------- softmax·v via WMMA f32 16x16x4 ----------------------------
  // D[u][0] = Σ_l E[u,l]·v[l]   D[u][1] = Σ_l E[u,l]   (B col0=v, col1=1)
  const int er  = tid & 15;           // E-tile row this thread fills
  const int ecb = (tid >> 4) << 3;    // E-tile col base (8 cols per thread)
  for (int t = 0; t < NT; ++t) {
    v8f acc = {0.f, 0.f, 0.f, 0.f, 0.f, 0.f, 0.f, 0.f};
    const float qu = sQs[t * 16 + er];
    const float mu = sMx[t * 16 + er];
    for (int l0 = 0; l0 < LL; l0 += LCH) {
      #pragma unroll
      for (int i = 0; i < 8; ++i) {
        int lc = ecb + i;
        float kl = fmaf(sX[l0 + lc], vWk, vbk);
        sE[er * SEP + lc] = expf(qu * kl - mu);
      }
      __syncthreads();
      if (wid == 0) {   // full wave, EXEC all ones
        const int half = lane >> 4;     // 0: K{0,1}, 1: K{2,3}
        const int mrow = lane & 15;     // A row / B column
        for (int kk = 0; kk < LCH; kk += 4) {
          v2f A;
          A.x = sE[mrow * SEP + kk + 2 * half];
          A.y = sE[mrow * SEP + kk + 2 * half + 1];
          float vv0 = fmaf(sX[l0 + kk + 2 * half],     vWv, vbv);
          float vv1 = fmaf(sX[l0 + kk + 2 * half + 1], vWv, vbv);
          v2f Bv;
          Bv.x = (mrow == 0) ? vv0 : ((mrow == 1) ? 1.0f : 0.0f);
          Bv.y = (mrow == 0) ? vv1 : ((mrow == 1) ? 1.0f : 0.0f);
          acc = __builtin_amdgcn_wmma_f32_16x16x4_f32(
                  false, A, false, Bv, (short)0, acc, false, false);
        }
      }
      __syncthreads();
    }
    if (wid == 0) {
      // D layout: VGPR r -> rows r (lanes 0-15) / r+8 (lanes 16-31), col = lane&15
      if (lane == 0) {
        for (int r = 0; r < 8; ++r) sNum[t * 16 + r] = acc[r];
      }
      if (lane == 16) {
        for (int r = 0; r < 8; ++r) sNum[t * 16 + 8 + r] = acc[r];
      }
      if (lane == 1) {
        for (int r = 0; r < 8; ++r) sDen[t * 16 + r] = acc[r];
      }
      if (lane == 17) {
        for (int r = 0; r < 8; ++r) sDen[t * 16 + 8 + r] = acc[r];
      }
    }
    __syncthreads();
  }

  // ------------- scatter upd into mean(v) background (reuse sM as ctx) -----
  #pragma unroll
  for (int i = 0; i < PERT; ++i) sM[tid + i * NTHR] = vmean;
  __syncthreads();
  if (tid < U) sM[sSel[tid]] = sNum[tid] / sDen[tid];
  __syncthreads();

  // ------------- residual + LN1 + exact GELU + LN2 -------------------------
  float tv[PERT];
  float ls = 0.f;
  #pragma unroll
  for (int i = 0; i < PERT; ++i) {
    int l = tid + i * NTHR;
    float tt = sX[l] + fmaf(sM[l], vWo, vbo);
    tv[i] = tt; ls += tt;
  }
  float mean1 = wg_reduce_sum(ls, lane, wid, sR) / (float)LL;
  float ls2 = 0.f;
  #pragma unroll
  for (int i = 0; i < PERT; ++i) { float d = tv[i] - mean1; ls2 += d * d; }
  float inv1 = rsqrtf(wg_reduce_sum(ls2, lane, wid, sR) / (float)LL + EPSV);

  ls = 0.f;
  #pragma unroll
  for (int i = 0; i < PERT; ++i) {
    float x1 = (tv[i] - mean1) * inv1 * vg1 + vbe1;
    float a  = fmaf(x1, vw1, vb1);
    float ge = 0.5f * a * (1.0f + erff(a * 0.70710678118654752f));
    tv[i] = x1 + fmaf(ge, vw2, vb2);
    ls += tv[i];
  }
  float mean2 = wg_reduce_sum(ls, lane, wid, sR) / (float)LL;
  ls2 = 0.f;
  #pragma unroll
  for (int i = 0; i < PERT; ++i) { float d = tv[i] - mean2; ls2 += d * d; }
  float inv2 = rsqrtf(wg_reduce_sum(ls2, lane, wid, sR) / (float)LL + EPSV);

  float* __restrict__ ob = out + (size_t)b * (LL * CC) + c;
  #pragma unroll
  for (int i = 0; i < PERT; ++i) {
    int l = tid + i * NTHR;
    ob[(size_t)l * CC] = (tv[i] - mean2) * inv2 * vg2 + vbe2;
  }
}

extern "C" void kernel_launch(void* const* d_in, const int* in_sizes, int n_in,
                              void* d_out, int out_size, void* d_ws, size_t ws_size,
                              hipStream_t stream) {
  (void)n_in; (void)out_size; (void)d_ws; (void)ws_size;
  const float* x   = (const float*)d_in[0];
  const float* Wq  = (const float*)d_in[1];
  const float* bq  = (const float*)d_in[2];
  const float* Wk  = (const float*)d_in[3];
  const float* bk  = (const float*)d_in[4];
  const float* Wv  = (const float*)d_in[5];
  const float* bv  = (const float*)d_in[6];
  const float* Wo  = (const float*)d_in[7];
  const float* bo  = (const float*)d_in[8];
  const float* w1  = (const float*)d_in[9];
  const float* b1  = (const float*)d_in[10];
  const float* w2  = (const float*)d_in[11];
  const float* b2  = (const float*)d_in[12];
  const float* g1  = (const float*)d_in[13];
  const float* be1 = (const float*)d_in[14];
  const float* g2  = (const float*)d_in[15];
  const float* be2 = (const float*)d_in[16];
  const int*   idx = (const int*)d_in[17];
  float* out = (float*)d_out;

  int U = in_sizes[17] / LL;       // = 40 for this config
  if (U > MAXU) U = MAXU;
  int NT = (U + 15) / 16;

  dim3 grid(BB * CC);
  dim3 block(NTHR);
  informer_fused<<<grid, block, 0, stream>>>(x, Wq, bq, Wk, bk, Wv, bv, Wo, bo,
                                             w1, b1, w2, b2, g1, be1, g2, be2,
                                             idx, out, U, NT);
}